// EMG_SyEMB_65979287601583
// MI455X (gfx1250) — compile-verified
//
#include <hip/hip_runtime.h>
#include <math.h>

// ---------------------------------------------------------------------------
// Problem constants (from reference)
// ---------------------------------------------------------------------------
#define BB     64
#define TT     50000
#define CC     4
#define MSYN   4
#define DD     160
#define PATCH  25
#define LL     2000            // T2 / PATCH = 50000/25
#define KENV   25
#define KBUR   9
#define KPRE   9

// front-end tiling
#define TC     1000            // time chunk per block (multiple of PATCH)
#define XOFF   17              // x halo: env needs +-12, s0 halo +-4, dx needs -1 -> 17
#define XN     (TC + 2*XOFF)   // 1034
#define S0OFF  4
#define S0N    (TC + 2*S0OFF)  // 1008

// GEMM tiling
#define MROWS  64              // rows per block (4 x 16-row WMMA tiles)
#define KTOT   100             // MSYN*PATCH
#define NWAVES 10              // 10 waves x 16 cols = 160 = D

typedef __attribute__((ext_vector_type(2))) float v2f;
typedef __attribute__((ext_vector_type(8))) float v8f;

__device__ __forceinline__ float gelu_exact(float x) {
    return 0.5f * x * (1.0f + erff(x * 0.70710678118654752f));
}

// ---------------------------------------------------------------------------
// Kernel 1: fused mask * convs * synergy mix * pre-stack * patch-scatter.
// One block handles one (batch, 1000-step chunk). Writes A[r, m*25+k] and
// m_patch.
// ---------------------------------------------------------------------------
__global__ __launch_bounds__(256)
void frontend_kernel(const float* __restrict__ X, const float* __restrict__ M,
                     const float* __restrict__ w_env, const float* __restrict__ w_burst,
                     const float* __restrict__ syn_raw, const float* __restrict__ w_dw,
                     const float* __restrict__ w_pw,
                     float* __restrict__ A, float* __restrict__ m_patch) {
    __shared__ float x_s[CC][XN + 6];      // masked x, halo included
    __shared__ float s0_s[MSYN][S0N + 8];  // synergy-mixed signal (pre dwconv)
    __shared__ float sm_s[MSYN][TC];       // clipped synergy mask
    __shared__ float mt_s[TC];             // time mask
    __shared__ float we_s[CC][KENV];
    __shared__ float wb_s[CC][KBUR];
    __shared__ float wsyn_s[MSYN][CC];
    __shared__ float wdw_s[MSYN][KPRE];
    __shared__ float wpw_s[MSYN][MSYN];

    const int tid   = threadIdx.x;
    const int b     = blockIdx.x / (TT / TC);
    const int chunk = blockIdx.x % (TT / TC);
    const int t0    = chunk * TC;          // chunk start within the batch

    // ---- small weights into LDS; softplus for synergy ----
    if (tid < CC * KENV)  we_s[tid / KENV][tid % KENV] = w_env[tid];
    if (tid < CC * KBUR)  wb_s[tid / KBUR][tid % KBUR] = w_burst[tid];
    if (tid < MSYN*KPRE)  wdw_s[tid / KPRE][tid % KPRE] = w_dw[tid];
    if (tid < MSYN*MSYN)  wpw_s[tid / MSYN][tid % MSYN] = w_pw[tid];
    if (tid < MSYN*CC) {
        float v  = syn_raw[tid];
        float sp = (v > 20.0f) ? v : log1pf(expf(v));   // softplus
        wsyn_s[tid / CC][tid % CC] = sp;
    }
    __syncthreads();
    if (tid < MSYN) {
        float s = wsyn_s[tid][0] + wsyn_s[tid][1] + wsyn_s[tid][2] + wsyn_s[tid][3];
        s = fmaxf(s, 1e-6f);
        float inv = 1.0f / s;
        wsyn_s[tid][0] *= inv; wsyn_s[tid][1] *= inv;
        wsyn_s[tid][2] *= inv; wsyn_s[tid][3] *= inv;
    }
    __syncthreads();

    // ---- load masked x (with halo); also mask mix for chunk region ----
    for (int i = tid; i < XN; i += 256) {
        int tg = t0 - XOFF + i;            // global t within batch
        float4 xv = make_float4(0.f, 0.f, 0.f, 0.f);
        float4 mv = make_float4(0.f, 0.f, 0.f, 0.f);
        if (tg >= 0 && tg < TT) {
            long base = ((long)b * TT + tg);
            xv = ((const float4*)X)[base];
            mv = ((const float4*)M)[base];
        }
        x_s[0][i] = xv.x * mv.x;
        x_s[1][i] = xv.y * mv.y;
        x_s[2][i] = xv.z * mv.z;
        x_s[3][i] = xv.w * mv.w;
        int tt = i - XOFF;
        if (tt >= 0 && tt < TC) {
            float mm[CC];
            mm[0] = (mv.x > 0.f) ? 1.f : 0.f;
            mm[1] = (mv.y > 0.f) ? 1.f : 0.f;
            mm[2] = (mv.z > 0.f) ? 1.f : 0.f;
            mm[3] = (mv.w > 0.f) ? 1.f : 0.f;
            #pragma unroll
            for (int m = 0; m < MSYN; ++m) {
                float s = wsyn_s[m][0]*mm[0] + wsyn_s[m][1]*mm[1]
                        + wsyn_s[m][2]*mm[2] + wsyn_s[m][3]*mm[3];
                sm_s[m][tt] = fminf(fmaxf(s, 0.0f), 1.0f);
            }
        }
    }
    __syncthreads();

    // ---- env/burst/xm and synergy mix -> s0 (with +-4 halo) ----
    for (int i = tid; i < S0N; i += 256) {
        int tg = t0 - S0OFF + i;           // global t of this s0 sample
        if (tg < 0 || tg >= TT) {          // reference zero-pads S before dwconv
            #pragma unroll
            for (int m = 0; m < MSYN; ++m) s0_s[m][i] = 0.0f;
            continue;
        }
        const int j = i + (XOFF - S0OFF);  // center index into x_s
        float xm[CC];
        #pragma unroll
        for (int c = 0; c < CC; ++c) {
            float env = 0.0f;
            #pragma unroll
            for (int k = 0; k < KENV; ++k)
                env += fabsf(x_s[c][j - 12 + k]) * we_s[c][k];
            float bur = 0.0f;
            #pragma unroll
            for (int k = 0; k < KBUR; ++k) {
                int u = j - 4 + k;
                // dx[t] = x[t]-x[t-1], dx[0] = 0 (front pad)
                float dx = ((t0 - XOFF + u) == 0) ? 0.0f : (x_s[c][u] - x_s[c][u - 1]);
                bur += fabsf(dx) * wb_s[c][k];
            }
            xm[c] = 0.9f * env + 0.6f * bur + 0.2f * x_s[c][j];
        }
        #pragma unroll
        for (int m = 0; m < MSYN; ++m) {
            s0_s[m][i] = wsyn_s[m][0]*xm[0] + wsyn_s[m][1]*xm[1]
                       + wsyn_s[m][2]*xm[2] + wsyn_s[m][3]*xm[3];
        }
    }
    __syncthreads();

    // ---- dwconv+gelu, pointwise+gelu, mask multiply, scatter to A ----
    for (int tt = tid; tt < TC; tt += 256) {
        float s1[MSYN];
        #pragma unroll
        for (int m = 0; m < MSYN; ++m) {
            float acc = 0.0f;
            #pragma unroll
            for (int k = 0; k < KPRE; ++k)
                acc += s0_s[m][tt + k] * wdw_s[m][k];   // tt+4-4+k
            s1[m] = gelu_exact(acc);
        }
        int t  = t0 + tt;
        int l  = t / PATCH;
        int kk = t - l * PATCH;
        long r = (long)b * LL + l;
        float* arow = A + r * KTOT + kk;
        #pragma unroll
        for (int o = 0; o < MSYN; ++o) {
            float v = wpw_s[o][0]*s1[0] + wpw_s[o][1]*s1[1]
                    + wpw_s[o][2]*s1[2] + wpw_s[o][3]*s1[3];
            arow[o * PATCH] = gelu_exact(v) * sm_s[o][tt];
        }
        float msum = sm_s[0][tt] + sm_s[1][tt] + sm_s[2][tt] + sm_s[3][tt];
        mt_s[tt] = (msum * 0.25f > 0.0f) ? 1.0f : 0.0f;
    }
    __syncthreads();

    // ---- patch mask ----
    for (int lp = tid; lp < TC / PATCH; lp += 256) {
        float s = 0.0f;
        #pragma unroll
        for (int k = 0; k < PATCH; ++k) s += mt_s[lp * PATCH + k];
        m_patch[(long)b * LL + chunk * (TC / PATCH) + lp] =
            (s * (1.0f / PATCH) > 0.1f) ? 1.0f : 0.0f;
    }
}

// ---------------------------------------------------------------------------
// Kernel 2: reorder w_proj (D,MSYN,PATCH) -> WpT[j=(m*25+k)][d] so the WMMA
// B-fragment loads are coalesced.
// ---------------------------------------------------------------------------
__global__ __launch_bounds__(256)
void wproj_transpose(const float* __restrict__ w_proj, float* __restrict__ WpT) {
    int i = blockIdx.x * 256 + threadIdx.x;
    if (i < DD * KTOT) {
        int d = i / KTOT, j = i - d * KTOT;
        WpT[j * DD + d] = w_proj[i];
    }
}

// ---------------------------------------------------------------------------
// Kernel 3: WMMA GEMM (128000 x 160 x 100, f32) + LayerNorm over D.
// 320 threads = 10 wave32s; each wave owns a 16-column N-tile; block owns 64
// rows (4 x 16-row tiles). 25 x v_wmma_f32_16x16x4_f32 per tile.
// ---------------------------------------------------------------------------
__global__ __launch_bounds__(320)
void gemm_ln_kernel(const float* __restrict__ A, const float* __restrict__ WpT,
                    const float* __restrict__ gamma, const float* __restrict__ beta,
                    float* __restrict__ out) {
    __shared__ float a_s[MROWS * KTOT];     // 25.6 KB
    __shared__ float h_s[MROWS * DD];       // 40.96 KB
    __shared__ float ps[5 * MROWS];
    __shared__ float pq[5 * MROWS];
    __shared__ float mu_s[MROWS];
    __shared__ float rs_s[MROWS];

    const int tid = threadIdx.x;
    const long r0 = (long)blockIdx.x * MROWS;

    // stage A tile (coalesced)
    for (int i = tid; i < MROWS * KTOT; i += 320)
        a_s[i] = A[r0 * KTOT + i];
    __syncthreads();

    const int wave = tid >> 5;
    const int lane = tid & 31;
    const int lrow = lane & 15;
    const int hi   = lane >> 4;           // 0: K pair {4s,4s+1}, 1: {4s+2,4s+3}
    const int n0   = wave * 16;
    const int bcol = n0 + lrow;

    #pragma unroll
    for (int rt = 0; rt < MROWS / 16; ++rt) {
        v8f acc = {0.f, 0.f, 0.f, 0.f, 0.f, 0.f, 0.f, 0.f};
        const int abase = (rt * 16 + lrow) * KTOT + 2 * hi;
        #pragma unroll
        for (int s = 0; s < KTOT / 4; ++s) {
            v2f a, bf;
            a.x  = a_s[abase + 4 * s];
            a.y  = a_s[abase + 4 * s + 1];
            bf.x = WpT[(4 * s + 2 * hi) * DD + bcol];
            bf.y = WpT[(4 * s + 2 * hi + 1) * DD + bcol];
            acc = __builtin_amdgcn_wmma_f32_16x16x4_f32(
                false, a, false, bf, (short)0, acc, false, false);
        }
        #pragma unroll
        for (int j = 0; j < 8; ++j) {
            int row = rt * 16 + j + hi * 8;   // C layout: VGPR j -> M=j / M=j+8
            h_s[row * DD + n0 + lrow] = acc[j];
        }
    }
    __syncthreads();

    // ---- LayerNorm over D=160 ----
    {
        int r = tid & 63;                 // 0..63
        int g = tid >> 6;                 // 0..4 -> 32 cols each
        float s = 0.f, q = 0.f;
        const float* hrow = h_s + r * DD + g * 32;
        #pragma unroll
        for (int c = 0; c < 32; ++c) { float v = hrow[c]; s += v; q += v * v; }
        ps[g * MROWS + r] = s;
        pq[g * MROWS + r] = q;
    }
    __syncthreads();
    if (tid < MROWS) {
        float s = 0.f, q = 0.f;
        #pragma unroll
        for (int g = 0; g < 5; ++g) { s += ps[g * MROWS + tid]; q += pq[g * MROWS + tid]; }
        float mu  = s * (1.0f / DD);
        float var = q * (1.0f / DD) - mu * mu;
        mu_s[tid] = mu;
        rs_s[tid] = rsqrtf(var + 1e-5f);
    }
    __syncthreads();
    for (int idx = tid; idx < MROWS * DD; idx += 320) {
        int rr = idx / DD, c = idx - rr * DD;
        out[(r0 + rr) * DD + c] =
            (h_s[idx] - mu_s[rr]) * rs_s[rr] * gamma[c] + beta[c];
    }
}

// ---------------------------------------------------------------------------
// Launch
// ---------------------------------------------------------------------------
extern "C" void kernel_launch(void* const* d_in, const int* in_sizes, int n_in,
                              void* d_out, int out_size, void* d_ws, size_t ws_size,
                              hipStream_t stream) {
    const float* X       = (const float*)d_in[0];
    const float* M       = (const float*)d_in[1];
    const float* w_env   = (const float*)d_in[2];
    const float* w_burst = (const float*)d_in[3];
    const float* syn_raw = (const float*)d_in[4];
    const float* w_dw    = (const float*)d_in[5];
    const float* w_pw    = (const float*)d_in[6];
    const float* w_proj  = (const float*)d_in[7];
    const float* gamma   = (const float*)d_in[8];
    const float* beta    = (const float*)d_in[9];

    float* out_h  = (float*)d_out;                      // (64, 2000, 160)
    float* out_mp = out_h + (long)BB * LL * DD;         // (64, 2000)

    float* A   = (float*)d_ws;                          // (128000, 100)
    float* WpT = A + (long)BB * LL * KTOT;              // (100, 160)

    wproj_transpose<<<(DD * KTOT + 255) / 256, 256, 0, stream>>>(w_proj, WpT);
    frontend_kernel<<<BB * (TT / TC), 256, 0, stream>>>(
        X, M, w_env, w_burst, syn_raw, w_dw, w_pw, A, out_mp);
    gemm_ln_kernel<<<(BB * LL) / MROWS, 320, 0, stream>>>(
        A, WpT, gamma, beta, out_h);
}